// MPS_21629455303212
// MI455X (gfx1250) — compile-verified
//
#include <hip/hip_runtime.h>

typedef __attribute__((ext_vector_type(2))) float v2f;
typedef __attribute__((ext_vector_type(8))) float v8f;

#define BATCH 128
#define SITES 256
#define BOND 64
#define OUT_DIM 10
#define HALF_SITES 128
#define LDS_STRIDE 68   // 64 + 4 floats padding -> conflict-free A reads

// ---------------------------------------------------------------------------
// Chain body (templated: left/right clones, zero runtime branching).
// Each wave32 independently evolves 16 batch rows through 128 sites.
// Site update S <- S + S·(x0·C0[s] + x1·C1[s]) is expressed as ONE K=128 GEMM
// by folding the feature dim into K:
//     A'[m, 2r+i] = x_i · A[m, r]        (x is per-lane: A-layout row = ln)
//     B'[2r+i, n] = cores[s][...][i]     (feat-innermost layout == K-expanded)
// so every B fragment is 2 contiguous floats (direct b64, no shuffles) and
// every A' fragment is one LDS scalar splat times the (x0,x1) pair (one
// v_pk_mul_f32). Identity term of M accumulates in-place via the WMMA C
// operand. The per-wave LDS tile round trip is the acc->A layout transpose.
// ---------------------------------------------------------------------------
template <bool LEFT>
__device__ __forceinline__ void chain_body(
    const float* __restrict__ input,   // [128,256,2]
    const float* __restrict__ cores,   // [256,64,64,2]
    const float* __restrict__ bvec,    // [64] boundary vector
    float* __restrict__ vout,          // [2][128][64]
    float* scr, int wave, int lane)
{
    const int lh = lane >> 4;          // lane half
    const int ln = lane & 15;
    const int kb = lh * 2;             // K sub-offset inside a K=4 chunk
    const int batchBase = wave * 16;

    // ---- loop-invariant per-lane offsets (float units); kt/nt -> imms ----
    const unsigned xOff  = (unsigned)((batchBase + ln) * (SITES * 2));
    const unsigned bOffL = (unsigned)(kb * BOND + ln * 2);           // + kt*256 + nt*32
    const unsigned bOffR = (unsigned)(ln * (BOND * 2) + kb);         // + nt*2048 + kt*4
    const unsigned aOff  = (unsigned)(ln * LDS_STRIDE + lh);         // + kt*2
    const unsigned ldsW  = (unsigned)(lh * (8 * LDS_STRIDE) + ln);   // + v*LDS_STRIDE + nt*16
    const unsigned outOff = (unsigned)((batchBase + 8 * lh) * BOND + ln); // + v*64 + nt*16

    // ---- uniform (SGPR) per-site base pointers, bumped scalar-side ----
    const float* sc = cores + (LEFT ? (size_t)0 : (size_t)(SITES - 1) * BOND * BOND * 2);
    const float* xp = input + (LEFT ? 0 : (SITES - 1) * 2);
    const int    scStep = LEFT ? (BOND * BOND * 2) : -(BOND * BOND * 2);

    // init state: every batch row starts as the boundary vector
    v8f S[4];
#pragma unroll
    for (int nt = 0; nt < 4; ++nt) {
        const float val = bvec[nt * 16 + ln];
#pragma unroll
        for (int v = 0; v < 8; ++v) S[nt][v] = val;
    }

    for (int step = 0; step < HALF_SITES; ++step) {
        // prefetch next site's core block
        if (step + 1 < HALF_SITES)
            __builtin_prefetch(sc + scStep + (unsigned)(lane * 256), 0, 1);

        // per-lane feature pair for this site (batch row = A-layout row = ln)
        const v2f xv = *(const v2f*)(xp + xOff);

        // spill S snapshot (acc layout) -> LDS tile [16][LDS_STRIDE]
#pragma unroll
        for (int nt = 0; nt < 4; ++nt)
#pragma unroll
            for (int v = 0; v < 8; ++v)
                scr[ldsW + (unsigned)(v * LDS_STRIDE + nt * 16)] = S[nt][v];
        __builtin_amdgcn_wave_barrier();
        asm volatile("" ::: "memory");

        // one K=128 GEMM in 32 K-chunks of 4, accumulating in-place into S
#pragma unroll
        for (int kt = 0; kt < 32; ++kt) {
            const float as = scr[aOff + (unsigned)(kt * 2)];  // A[m][kt*2+lh]
            v2f a; a.x = as; a.y = as;
            const v2f af = a * xv;                            // (x0*as, x1*as)
#pragma unroll
            for (int nt = 0; nt < 4; ++nt) {
                const v2f b = LEFT
                    ? *(const v2f*)(sc + bOffL + (unsigned)(kt * 256 + nt * 32))
                    : *(const v2f*)(sc + bOffR + (unsigned)(nt * 2048 + kt * 4));
                S[nt] = __builtin_amdgcn_wmma_f32_16x16x4_f32(
                    false, af, false, b, (short)0, S[nt], false, false);
            }
        }

        sc += scStep;
        xp += LEFT ? 2 : -2;
    }

    // write propagated boundary vectors: vL at [0], vR at [BATCH*BOND]
    float* dst = vout + (LEFT ? (size_t)0 : (size_t)BATCH * BOND);
#pragma unroll
    for (int nt = 0; nt < 4; ++nt)
#pragma unroll
        for (int v = 0; v < 8; ++v)
            dst[outOff + (unsigned)(v * BOND + nt * 16)] = S[nt][v];
}

// ---------------------------------------------------------------------------
__launch_bounds__(256)
__global__ void mps_chain_kernel(const float* __restrict__ input,
                                 const float* __restrict__ cores,
                                 const float* __restrict__ lvec,
                                 const float* __restrict__ rvec,
                                 float* __restrict__ vout)
{
    __shared__ float lds[8 * 16 * LDS_STRIDE];
    const int wave = threadIdx.x >> 5;
    const int lane = threadIdx.x & 31;
    float* scr = lds + wave * (16 * LDS_STRIDE);
    if (blockIdx.x == 0)
        chain_body<true>(input, cores, lvec, vout, scr, wave, lane);
    else
        chain_body<false>(input, cores, rvec, vout, scr, wave, lane);
}

// ---------------------------------------------------------------------------
// Combine: logits[b,o] = sum_{l,r} vL[b,l] * out_core[o,l,r] * vR[b,r]
// One wave32 per (b,o); lane handles r and r+32, then shuffle-reduce.
// ---------------------------------------------------------------------------
__launch_bounds__(32)
__global__ void mps_combine_kernel(const float* __restrict__ vL,       // [128,64]
                                   const float* __restrict__ vR,       // [128,64]
                                   const float* __restrict__ out_core, // [10,64,64]
                                   float* __restrict__ logits)         // [128,10]
{
    const int b    = blockIdx.x;
    const int o    = blockIdx.y;
    const int lane = threadIdx.x;
    const float* l  = vL + (size_t)b * BOND;
    const float* r  = vR + (size_t)b * BOND;
    const float* oc = out_core + (size_t)o * BOND * BOND;

    float accA = 0.0f, accB = 0.0f;
    const int rA = lane, rB = lane + 32;
#pragma unroll 8
    for (int j = 0; j < BOND; ++j) {
        const float lv = l[j];
        accA = fmaf(lv, oc[j * BOND + rA], accA);
        accB = fmaf(lv, oc[j * BOND + rB], accB);
    }
    float part = fmaf(accA, r[rA], accB * r[rB]);
#pragma unroll
    for (int off = 16; off > 0; off >>= 1)
        part += __shfl_xor(part, off, 32);
    if (lane == 0) logits[(size_t)b * OUT_DIM + o] = part;
}

// ---------------------------------------------------------------------------
extern "C" void kernel_launch(void* const* d_in, const int* in_sizes, int n_in,
                              void* d_out, int out_size, void* d_ws, size_t ws_size,
                              hipStream_t stream) {
    const float* input    = (const float*)d_in[0];  // [128,256,2]
    const float* cores    = (const float*)d_in[1];  // [256,64,64,2]
    const float* out_core = (const float*)d_in[2];  // [10,64,64]
    const float* lvec     = (const float*)d_in[3];  // [64]
    const float* rvec     = (const float*)d_in[4];  // [64]
    float* logits = (float*)d_out;                  // [128,10]
    float* vbuf   = (float*)d_ws;                   // [2][128][64] scratch

    mps_chain_kernel<<<dim3(2), dim3(256), 0, stream>>>(input, cores, lvec, rvec, vbuf);
    mps_combine_kernel<<<dim3(BATCH, OUT_DIM), dim3(32), 0, stream>>>(
        vbuf, vbuf + (size_t)BATCH * BOND, out_core, logits);
}